// InnerProductLayer_20126216749638
// MI455X (gfx1250) — compile-verified
//
#include <hip/hip_runtime.h>

typedef __attribute__((ext_vector_type(2))) float v2f;
typedef __attribute__((ext_vector_type(8))) float v8f;

#define NFIELDS 32
#define DDIM    64
#define PAIRS   (NFIELDS * (NFIELDS - 1) / 2)   // 496
#define WAVES_PER_BLOCK 8

// One wave32 computes one batch's 32x32 Gram matrix via fp32 WMMA.
// Tiles computed: (0,0), (0,1), (1,1) -- tile (1,0) is the transpose of (0,1).
// A(16x4 f32) per-lane layout == B(4x16 f32) per-lane layout as a function of
// (lane, vgpr), so the same two half-rows per lane feed both operands.
__global__ void __launch_bounds__(256)
inner_product_wmma(const float* __restrict__ x, float* __restrict__ out, int nbatch) {
    __shared__ float gram[WAVES_PER_BLOCK][NFIELDS * NFIELDS];  // 8 * 4KB = 32KB

    const int tid  = threadIdx.x;
    const int lane = tid & 31;
    const int wave = tid >> 5;
    const int b    = blockIdx.x * WAVES_PER_BLOCK + wave;   // wave-uniform

    const int r  = lane & 15;   // row within 16-row tile / N column within tile
    const int hi = lane >> 4;   // which K-pair of the 4-wide K slice

    if (b < nbatch) {
        const float* xb   = x + (size_t)b * (NFIELDS * DDIM);
        const float* p_lo = xb + r * DDIM + 2 * hi;        // row r, its K-pair half
        const float* p_hi = p_lo + 16 * DDIM;              // row r+16, same half

        // Each lane: 16 contiguous 8B loads per row-half; whole 8KB batch is
        // loaded exactly once across the wave.
        v2f row_lo[16], row_hi[16];
        #pragma unroll
        for (int k = 0; k < 16; ++k) {
            row_lo[k] = *(const v2f*)(p_lo + 4 * k);
            row_hi[k] = *(const v2f*)(p_hi + 4 * k);
        }

        v8f acc00 = {0.f, 0.f, 0.f, 0.f, 0.f, 0.f, 0.f, 0.f};
        v8f acc01 = acc00;
        v8f acc11 = acc00;

        // 16 K-steps of 4 -> K=64; 48 x v_wmma_f32_16x16x4_f32 per batch.
        #pragma unroll
        for (int k = 0; k < 16; ++k) {
            acc00 = __builtin_amdgcn_wmma_f32_16x16x4_f32(
                false, row_lo[k], false, row_lo[k], (short)0, acc00, false, false);
            acc01 = __builtin_amdgcn_wmma_f32_16x16x4_f32(
                false, row_lo[k], false, row_hi[k], (short)0, acc01, false, false);
            acc11 = __builtin_amdgcn_wmma_f32_16x16x4_f32(
                false, row_hi[k], false, row_hi[k], (short)0, acc11, false, false);
        }

        // C/D layout: VGPR v, lanes 0-15 -> M=v, lanes 16-31 -> M=8+v; N = lane&15.
        float* g = gram[wave];
        #pragma unroll
        for (int v = 0; v < 8; ++v) {
            const int i0 = 8 * hi + v;
            g[i0 * NFIELDS + r]              = acc00[v];
            g[i0 * NFIELDS + (r + 16)]       = acc01[v];
            g[(i0 + 16) * NFIELDS + (r + 16)] = acc11[v];
        }
    }
    __syncthreads();

    if (b < nbatch) {
        const float* g  = gram[wave];
        float*       ob = out + (size_t)b * PAIRS;
        // p enumerates (i<j) pairs in row-major order, matching triu_indices(k=1).
        // Monotonic incremental decode of (i, j) from p; coalesced NT stores.
        int i = 0, base = 0, cnt = NFIELDS - 1;
        #pragma unroll
        for (int t = 0; t < (PAIRS + 31) / 32; ++t) {
            const int p = lane + 32 * t;
            if (p < PAIRS) {
                while (p >= base + cnt) { base += cnt; --cnt; ++i; }
                const int j = i + 1 + (p - base);
                __builtin_nontemporal_store(g[i * NFIELDS + j], ob + p);
            }
        }
    }
}

extern "C" void kernel_launch(void* const* d_in, const int* in_sizes, int n_in,
                              void* d_out, int out_size, void* d_ws, size_t ws_size,
                              hipStream_t stream) {
    const float* x   = (const float*)d_in[0];
    float*       out = (float*)d_out;
    const int nbatch = in_sizes[0] / (NFIELDS * DDIM);   // 16384
    const int blocks = (nbatch + WAVES_PER_BLOCK - 1) / WAVES_PER_BLOCK;
    inner_product_wmma<<<blocks, 256, 0, stream>>>(x, out, nbatch);
}